// CFLP_49082886258723
// MI455X (gfx1250) — compile-verified
//
#include <hip/hip_runtime.h>
#include <hip/hip_bf16.h>
#include <math.h>

// ---------------------------------------------------------------------------
// CFLP GAT + decoder for MI455X (gfx1250, wave32).
// GEMMs use V_WMMA_F32_16X16X4_F32 (fp32 WMMA path). Edge segment ops use
// native f32 global atomics (encoded-int atomicMax for segment_max).
// ---------------------------------------------------------------------------

#define N_NODES 50000
#define N_EDGES 800000
#define N_EL    131072
#define DZ      96

typedef __attribute__((ext_vector_type(2))) float v2f;
typedef __attribute__((ext_vector_type(8))) float v8f;

// order-preserving float<->int encoding for atomic max over signed floats
__device__ __forceinline__ int enc_f(float f) {
    int i = __float_as_int(f);
    return i >= 0 ? i : (i ^ 0x7fffffff);
}
__device__ __forceinline__ float dec_f(int i) {
    return __int_as_float(i >= 0 ? i : (i ^ 0x7fffffff));
}

// ---------------------------------------------------------------------------
// hl = x@Wl + bl ; hr = x@Wr + br   via V_WMMA_F32_16X16X4_F32
// One wave computes one 16x16 output tile. n is a multiple of 16 (50000).
// ---------------------------------------------------------------------------
__global__ void gat_gemm_wmma(const float* __restrict__ x,
                              const float* __restrict__ Wl, const float* __restrict__ bl,
                              const float* __restrict__ Wr, const float* __restrict__ br,
                              float* __restrict__ hl, float* __restrict__ hr,
                              int n, int di) {
    const int DO = DZ;                       // 96
    const int wavesPerBlock = blockDim.x >> 5;
    const int lane   = threadIdx.x & 31;
    const int waveId = blockIdx.x * wavesPerBlock + (threadIdx.x >> 5);
    const int stride = gridDim.x * wavesPerBlock;

    const int nRowTiles   = n >> 4;          // 3125
    const int nColTiles   = DO >> 4;         // 6
    const int unitsPerMat = nRowTiles * nColTiles;
    const int totalUnits  = unitsPerMat * 2;

    const int mrow  = lane & 15;
    const int khalf = (lane >> 4) * 2;       // 0 for lanes 0-15, 2 for 16-31

    for (int unit = waveId; unit < totalUnits; unit += stride) {
        int mat = unit / unitsPerMat;        // 0 -> Wl, 1 -> Wr
        int t   = unit - mat * unitsPerMat;
        int rowTile = t / nColTiles;
        int colTile = t - rowTile * nColTiles;
        const float* W    = mat ? Wr : Wl;
        const float* bias = mat ? br : bl;
        float*       out  = mat ? hr : hl;

        int row0 = rowTile << 4;
        int col0 = colTile << 4;
        int colIdx = col0 + mrow;

        const float* aRow = x + (size_t)(row0 + mrow) * di + khalf;
        const float* bCol = W + (size_t)khalf * DO + colIdx;

        v8f c = {};
        for (int k = 0; k < di; k += 4) {
            v2f a, b;
            a.x = aRow[k];
            a.y = aRow[k + 1];
            b.x = bCol[(size_t)k * DO];
            b.y = bCol[(size_t)k * DO + DO];
            c = __builtin_amdgcn_wmma_f32_16x16x4_f32(
                    /*neg_a=*/false, a, /*neg_b=*/false, b,
                    /*c_mod=*/(short)0, c, /*reuse_a=*/false, /*reuse_b=*/false);
        }

        float bv = bias[colIdx];
        int rbase = row0 + ((lane >> 4) << 3);   // rows 0..7 or 8..15 of tile
        #pragma unroll
        for (int r = 0; r < 8; ++r)
            out[(size_t)(rbase + r) * DO + colIdx] = c[r] + bv;
    }
}

// ---------------------------------------------------------------------------
// init per-layer accumulators: m = enc(-inf), s = 0, agg = 0, stats = 0
// ---------------------------------------------------------------------------
__global__ void gat_init(int* __restrict__ menc, float* __restrict__ ssum,
                         float* __restrict__ agg, float* __restrict__ stats, int n) {
    size_t total = (size_t)n * DZ;
    int negInf = enc_f(-INFINITY);
    for (size_t i = (size_t)blockIdx.x * blockDim.x + threadIdx.x; i < total;
         i += (size_t)gridDim.x * blockDim.x) {
        menc[i] = negInf;
        ssum[i] = 0.0f;
        agg[i]  = 0.0f;
        if (i < 2 * DZ) stats[i] = 0.0f;
    }
}

__device__ __forceinline__ float edge_e(const float* __restrict__ hl,
                                        const float* __restrict__ hr,
                                        const float* __restrict__ al,
                                        const float* __restrict__ ar,
                                        int s, int d, int k) {
    float v = ar[k] * hr[(size_t)d * DZ + k] + al[k] * hl[(size_t)s * DZ + k];
    return v >= 0.0f ? v : 0.01f * v;        // leaky_relu, slope 0.01
}

// pass A: segment max of e over dst
__global__ void gat_edge_max(const int* __restrict__ src, const int* __restrict__ dst,
                             const float* __restrict__ hl, const float* __restrict__ hr,
                             const float* __restrict__ al, const float* __restrict__ ar,
                             int* __restrict__ menc, int E) {
    int k = threadIdx.x;                      // blockDim.x == 96
    for (int e = blockIdx.x; e < E; e += gridDim.x) {
        int s = src[e], d = dst[e];
        float ev = edge_e(hl, hr, al, ar, s, d, k);
        atomicMax(&menc[(size_t)d * DZ + k], enc_f(ev));
    }
}

// pass B: segment sum of exp(e - m[dst])
__global__ void gat_edge_sum(const int* __restrict__ src, const int* __restrict__ dst,
                             const float* __restrict__ hl, const float* __restrict__ hr,
                             const float* __restrict__ al, const float* __restrict__ ar,
                             const int* __restrict__ menc, float* __restrict__ ssum, int E) {
    int k = threadIdx.x;
    for (int e = blockIdx.x; e < E; e += gridDim.x) {
        int s = src[e], d = dst[e];
        float ev = edge_e(hl, hr, al, ar, s, d, k);
        float ex = expf(ev - dec_f(menc[(size_t)d * DZ + k]));
        atomicAdd(&ssum[(size_t)d * DZ + k], ex);
    }
}

// pass C: agg[dst] += (ex / (s[dst]+1e-16)) * hr[src]
__global__ void gat_edge_agg(const int* __restrict__ src, const int* __restrict__ dst,
                             const float* __restrict__ hl, const float* __restrict__ hr,
                             const float* __restrict__ al, const float* __restrict__ ar,
                             const int* __restrict__ menc, const float* __restrict__ ssum,
                             float* __restrict__ agg, int E) {
    int k = threadIdx.x;
    for (int e = blockIdx.x; e < E; e += gridDim.x) {
        int s = src[e], d = dst[e];
        size_t di = (size_t)d * DZ + k;
        float ev = edge_e(hl, hr, al, ar, s, d, k);
        float ex = expf(ev - dec_f(menc[di]));
        float a  = ex / (ssum[di] + 1e-16f);
        atomicAdd(&agg[di], a * hr[(size_t)s * DZ + k]);
    }
}

// ---------------------------------------------------------------------------
// batchnorm: per-feature sum / sumsq reduction, then normalize + ELU
// ---------------------------------------------------------------------------
__global__ void bn_reduce(const float* __restrict__ x, float* __restrict__ stats, int n) {
    __shared__ float ls[DZ];
    __shared__ float lq[DZ];
    if (threadIdx.x < DZ) { ls[threadIdx.x] = 0.0f; lq[threadIdx.x] = 0.0f; }
    __syncthreads();
    size_t total = (size_t)n * DZ;
    for (size_t i = (size_t)blockIdx.x * blockDim.x + threadIdx.x; i < total;
         i += (size_t)gridDim.x * blockDim.x) {
        int k = (int)(i % DZ);
        float v = x[i];
        atomicAdd(&ls[k], v);
        atomicAdd(&lq[k], v * v);
    }
    __syncthreads();
    if (threadIdx.x < DZ) {
        atomicAdd(&stats[threadIdx.x],      ls[threadIdx.x]);
        atomicAdd(&stats[DZ + threadIdx.x], lq[threadIdx.x]);
    }
}

__global__ void bn_apply_elu(const float* __restrict__ x, const float* __restrict__ stats,
                             const float* __restrict__ gamma, const float* __restrict__ beta,
                             float* __restrict__ out, int n) {
    size_t total = (size_t)n * DZ;
    float invN = 1.0f / (float)n;
    for (size_t i = (size_t)blockIdx.x * blockDim.x + threadIdx.x; i < total;
         i += (size_t)gridDim.x * blockDim.x) {
        int k = (int)(i % DZ);
        float mu  = stats[k] * invN;
        float var = stats[DZ + k] * invN - mu * mu;
        float y = gamma[k] * (x[i] - mu) * rsqrtf(var + 1e-5f) + beta[k];
        out[i] = y > 0.0f ? y : expm1f(y);
    }
}

// ---------------------------------------------------------------------------
// jumping knowledge: z = sum_i softmax(jk_w)[i] * out_i
// ---------------------------------------------------------------------------
__global__ void jk_combine(const float* __restrict__ o0, const float* __restrict__ o1,
                           const float* __restrict__ o2, const float* __restrict__ jkw,
                           float* __restrict__ z, int n) {
    float w0 = jkw[0], w1 = jkw[1], w2 = jkw[2];
    float mx = fmaxf(w0, fmaxf(w1, w2));
    float e0 = expf(w0 - mx), e1 = expf(w1 - mx), e2 = expf(w2 - mx);
    float inv = 1.0f / (e0 + e1 + e2);
    e0 *= inv; e1 *= inv; e2 *= inv;
    size_t total = (size_t)n * DZ;
    for (size_t i = (size_t)blockIdx.x * blockDim.x + threadIdx.x; i < total;
         i += (size_t)gridDim.x * blockDim.x) {
        z[i] = e0 * o0[i] + e1 * o1[i] + e2 * o2[i];
    }
}

// ---------------------------------------------------------------------------
// decoder: h = elu([z_i*z_j, T] @ W1 + b1); out = h @ W2
// blockDim = 96, one edge-task at a time per block, W1/W2 staged in LDS and
// amortized across grid-stride tasks. task < EL -> T_f batch, else T_cf.
// ---------------------------------------------------------------------------
__global__ void decode_kernel(const float* __restrict__ z, const int* __restrict__ edges,
                              const float* __restrict__ Tf, const float* __restrict__ Tcf,
                              const float* __restrict__ W1, const float* __restrict__ b1,
                              const float* __restrict__ W2,
                              float* __restrict__ outF, float* __restrict__ outCF, int EL) {
    __shared__ float W1s[97 * DZ];
    __shared__ float W2s[DZ];
    __shared__ float prod[97];
    __shared__ float red[DZ];

    for (int i = threadIdx.x; i < 97 * DZ; i += blockDim.x) W1s[i] = W1[i];
    if (threadIdx.x < DZ) W2s[threadIdx.x] = W2[threadIdx.x];
    __syncthreads();

    int tid = threadIdx.x;                   // 0..95
    int totalTasks = 2 * EL;
    for (int task = blockIdx.x; task < totalTasks; task += gridDim.x) {
        int b = task / EL;                   // uniform per block
        int e = task - b * EL;
        int vi = edges[2 * e + 0];
        int vj = edges[2 * e + 1];
        prod[tid] = z[(size_t)vi * DZ + tid] * z[(size_t)vj * DZ + tid];
        if (tid == 0) prod[DZ] = (b ? Tcf : Tf)[e];
        __syncthreads();

        float acc = b1[tid];
        #pragma unroll 1
        for (int k = 0; k < 97; ++k)
            acc = fmaf(prod[k], W1s[k * DZ + tid], acc);
        acc = acc > 0.0f ? acc : expm1f(acc);
        red[tid] = acc * W2s[tid];
        __syncthreads();

        if (tid < 48) red[tid] += red[tid + 48];
        __syncthreads();
        if (tid < 24) red[tid] += red[tid + 24];
        __syncthreads();
        if (tid < 12) red[tid] += red[tid + 12];
        __syncthreads();
        if (tid < 6)  red[tid] += red[tid + 6];
        __syncthreads();
        if (tid == 0) {
            float r = red[0] + red[1] + red[2] + red[3] + red[4] + red[5];
            (b ? outCF : outF)[e] = r;
        }
        __syncthreads();
    }
}

// ---------------------------------------------------------------------------
// launch
// ---------------------------------------------------------------------------
extern "C" void kernel_launch(void* const* d_in, const int* in_sizes, int n_in,
                              void* d_out, int out_size, void* d_ws, size_t ws_size,
                              hipStream_t stream) {
    const int   N  = N_NODES;
    const int   E  = N_EDGES;
    const int   EL = N_EL;

    const float* features   = (const float*)d_in[0];
    const int*   edge_index = (const int*)  d_in[1];
    const int*   edges      = (const int*)  d_in[2];
    const float* Tf         = (const float*)d_in[3];
    const float* Tcf        = (const float*)d_in[4];
    // params flattened in dict insertion order starting at d_in[5]:
    // per layer l: Wl, bl, Wr, br, al, ar, gamma, beta  (8 each, 3 layers)
    // then jk_w, W1, b1, W2
    auto P = [&](int i) { return (const float*)d_in[5 + i]; };

    const int* src = edge_index;         // edge_index[0]
    const int* dst = edge_index + E;     // edge_index[1]

    // workspace layout (floats): hl, hr, m_enc(int), s, agg, out0, out1, out2, stats
    size_t NF = (size_t)N * DZ;
    float* ws     = (float*)d_ws;
    float* hl     = ws;
    float* hr     = ws + NF;
    int*   menc   = (int*)(ws + 2 * NF);
    float* ssum   = ws + 3 * NF;
    float* agg    = ws + 4 * NF;
    float* outl0  = ws + 5 * NF;
    float* outl1  = ws + 6 * NF;
    float* outl2  = ws + 7 * NF;
    float* stats  = ws + 8 * NF;
    float* outl[3] = {outl0, outl1, outl2};

    float* z     = (float*)d_out;        // (N, 96)
    float* decF  = z + NF;               // (EL,)
    float* decCF = decF + EL;            // (EL,)

    int elemBlocks = (int)((NF + 255) / 256);

    const float* xin = features;
    int di = 128;
    for (int l = 0; l < 3; ++l) {
        const float* Wl    = P(8 * l + 0);
        const float* bl    = P(8 * l + 1);
        const float* Wr    = P(8 * l + 2);
        const float* br    = P(8 * l + 3);
        const float* al    = P(8 * l + 4);
        const float* ar    = P(8 * l + 5);
        const float* gamma = P(8 * l + 6);
        const float* beta  = P(8 * l + 7);

        gat_gemm_wmma<<<1024, 256, 0, stream>>>(xin, Wl, bl, Wr, br, hl, hr, N, di);
        gat_init<<<elemBlocks, 256, 0, stream>>>(menc, ssum, agg, stats, N);
        gat_edge_max<<<E, DZ, 0, stream>>>(src, dst, hl, hr, al, ar, menc, E);
        gat_edge_sum<<<E, DZ, 0, stream>>>(src, dst, hl, hr, al, ar, menc, ssum, E);
        gat_edge_agg<<<E, DZ, 0, stream>>>(src, dst, hl, hr, al, ar, menc, ssum, agg, E);
        bn_reduce<<<4096, 256, 0, stream>>>(agg, stats, N);
        bn_apply_elu<<<elemBlocks, 256, 0, stream>>>(agg, stats, gamma, beta, outl[l], N);

        xin = outl[l];
        di  = DZ;
    }

    jk_combine<<<elemBlocks, 256, 0, stream>>>(outl0, outl1, outl2, P(24), z, N);
    decode_kernel<<<8192, DZ, 0, stream>>>(z, edges, Tf, Tcf, P(25), P(26), P(27),
                                           decF, decCF, EL);
}